// SmoothnessLoss_4587025072325
// MI455X (gfx1250) — compile-verified
//
#include <hip/hip_runtime.h>
#include <hip/hip_bf16.h>
#include <math.h>

// ---------------------------------------------------------------------------
// SmoothnessLoss for MI455X (gfx1250).
// Memory-bound stencil+transcendental reduction: WMMA is inapplicable (no
// contraction). CDNA5 feature used: async global->LDS copies (ASYNCcnt path)
// for halo-tile staging, LDS-resident 3x3 stencil math, deterministic
// two-stage reduction (graph-replay safe, no float atomics).
// ---------------------------------------------------------------------------

#define TW 32
#define TH 8
#define HALO_W (TW + 2)          // 34
#define HALO_H (TH + 2)          // 10
#define PLANE (HALO_W * HALO_H)  // 340
#define NPLANES 6                // 3 orig channels + 3 smooth channels
#define IMG_H 512
#define IMG_W 512
#define EPSV 1e-8f
#define LOG2E 1.4426950408889634f

#if __has_builtin(__builtin_amdgcn_global_load_async_to_lds_b32)
#define ASYNC_LDS 1
#else
#define ASYNC_LDS 0
#endif

typedef __attribute__((address_space(1))) int as1_int;
typedef __attribute__((address_space(3))) int as3_int;

__device__ __forceinline__ void lds_copy_f32(const float* g, float* l) {
#if ASYNC_LDS
  // emits global_load_async_to_lds_b32 (tracked by ASYNCcnt)
  __builtin_amdgcn_global_load_async_to_lds_b32(
      (as1_int*)(g), (as3_int*)(l), /*offset=*/0, /*cpol=*/0);
#else
  *l = *g;
#endif
}

__device__ __forceinline__ void wait_async_lds() {
#if ASYNC_LDS
#if __has_builtin(__builtin_amdgcn_s_wait_asynccnt)
  __builtin_amdgcn_s_wait_asynccnt(0);
#else
  asm volatile("s_wait_asynccnt 0" ::: "memory");
#endif
#endif
}

// hardware log2 (v_log_f32)
__device__ __forceinline__ float hw_log2(float x) {
#if __has_builtin(__builtin_amdgcn_logf)
  return __builtin_amdgcn_logf(x);
#else
  return log2f(x);
#endif
}

// hardware exp2 (v_exp_f32)
__device__ __forceinline__ float hw_exp2(float x) {
#if __has_builtin(__builtin_amdgcn_exp2f)
  return __builtin_amdgcn_exp2f(x);
#else
  return exp2f(x);
#endif
}

__device__ __forceinline__ int reflect_idx(int v, int n) {
  // jnp.pad mode='reflect' with pad 1: -1 -> 1, n -> n-2
  v = (v < 0) ? -v : v;
  v = (v >= n) ? (2 * n - 2 - v) : v;
  return v;
}

__global__ __launch_bounds__(256) void smooth_loss_kernel(
    const float* __restrict__ orig, const float* __restrict__ smooth,
    float* __restrict__ partial) {
  __shared__ float lds[NPLANES * PLANE];
  __shared__ float red[256];

  const int tx = threadIdx.x;          // [0,32)
  const int ty = threadIdx.y;          // [0,8)
  const int tid = ty * TW + tx;
  const int x0 = blockIdx.x * TW;
  const int y0 = blockIdx.y * TH;
  const int b = blockIdx.z;
  const size_t img_base = (size_t)b * 3 * IMG_H * IMG_W;

  // ---- stage halo tiles (both images, 3 channels) into LDS, async ----
  for (int idx = tid; idx < NPLANES * PLANE; idx += 256) {
    int plane = idx / PLANE;
    int rem = idx - plane * PLANE;
    int i = rem / HALO_W;
    int j = rem - i * HALO_W;
    int gy = reflect_idx(y0 + i - 1, IMG_H);
    int gx = reflect_idx(x0 + j - 1, IMG_W);
    const float* src = (plane < 3) ? orig : smooth;
    int c = (plane < 3) ? plane : (plane - 3);
    lds_copy_f32(&src[img_base + ((size_t)c * IMG_H + gy) * IMG_W + gx],
                 &lds[idx]);
  }
  wait_async_lds();
  __syncthreads();

  const int ci = ty + 1;
  const int cj = tx + 1;
#define OO(c, i, j) lds[(c) * PLANE + (i) * HALO_W + (j)]
#define SS(c, i, j) lds[(3 + (c)) * PLANE + (i) * HALO_W + (j)]

  float oc[3], sc[3];
#pragma unroll
  for (int c = 0; c < 3; ++c) {
    oc[c] = OO(c, ci, cj);
    sc[c] = SS(c, ci, cj);
  }

  // ---- edge responses (gradient magnitude, summed over channels) ----
  float eo = 0.f, es = 0.f;
#pragma unroll
  for (int c = 0; c < 3; ++c) {
    float dxo = oc[c] - OO(c, ci + 1, cj);
    float dyo = oc[c] - OO(c, ci, cj + 1);
    eo += dxo * dxo + dyo * dyo;
    float dxs = sc[c] - SS(c, ci + 1, cj);
    float dys = sc[c] - SS(c, ci, cj + 1);
    es += dxs * dxs + dys * dys;
  }
  const bool use_large = (eo < 1.0f) && ((es - eo) > 1.0f);

  // spatial gaussian weights exp(-0.5*(dx^2+dy^2)) indexed by dx^2+dy^2
  const float ws_tab[3] = {1.0f, 0.60653065971263342f, 0.36787944117144233f};

  float acc = 0.f;
  if (use_large) {
    // p_large = 2 branch: ws[k] * (d+eps)^2, no transcendentals needed
#pragma unroll
    for (int oy = -1; oy <= 1; ++oy) {
#pragma unroll
      for (int ox = -1; ox <= 1; ++ox) {
        const float wsk = ws_tab[ox * ox + oy * oy];
#pragma unroll
        for (int c = 0; c < 3; ++c) {
          float dv = fabsf(sc[c] - SS(c, ci + ox, cj + oy)) + EPSV;
          acc += wsk * dv * dv;
        }
      }
    }
  } else {
    // p_small = 0.8 branch: wr[k] * (d+eps)^0.8 with bilateral color weight
#pragma unroll
    for (int oy = -1; oy <= 1; ++oy) {
#pragma unroll
      for (int ox = -1; ox <= 1; ++ox) {
        float so = 0.f, tsum = 0.f;
#pragma unroll
        for (int c = 0; c < 3; ++c) {
          float dob = oc[c] - OO(c, ci + ox, cj + oy);
          so += dob * dob;
          float dv = fabsf(sc[c] - SS(c, ci + ox, cj + oy)) + EPSV;
          // (d+eps)^0.8 = exp2(0.8 * log2(d+eps)) via v_log_f32/v_exp_f32
          tsum += hw_exp2(0.8f * hw_log2(dv));
        }
        // wr = exp(-0.5*so) = exp2(-0.5*log2(e)*so)
        acc += hw_exp2(-0.5f * LOG2E * so) * tsum;
      }
    }
  }
#undef OO
#undef SS

  // ---- block tree reduction (deterministic) ----
  red[tid] = acc;
  __syncthreads();
#pragma unroll
  for (int off = 128; off > 0; off >>= 1) {
    if (tid < off) red[tid] += red[tid + off];
    __syncthreads();
  }
  if (tid == 0) {
    partial[((size_t)b * gridDim.y + blockIdx.y) * gridDim.x + blockIdx.x] =
        red[0];
  }
}

__global__ __launch_bounds__(256) void reduce_partials_kernel(
    const float* __restrict__ partial, int n, float* __restrict__ out,
    double inv_n) {
  __shared__ double red[256];
  double s = 0.0;
  for (int i = threadIdx.x; i < n; i += 256) s += (double)partial[i];
  red[threadIdx.x] = s;
  __syncthreads();
#pragma unroll
  for (int off = 128; off > 0; off >>= 1) {
    if (threadIdx.x < off) red[threadIdx.x] += red[threadIdx.x + off];
    __syncthreads();
  }
  if (threadIdx.x == 0) out[0] = (float)(red[0] * inv_n);
}

extern "C" void kernel_launch(void* const* d_in, const int* in_sizes, int n_in,
                              void* d_out, int out_size, void* d_ws,
                              size_t ws_size, hipStream_t stream) {
  const float* orig = (const float*)d_in[0];    // [B,3,512,512] fp32
  const float* smooth = (const float*)d_in[1];  // [B,3,512,512] fp32
  float* out = (float*)d_out;                   // scalar fp32
  float* partial = (float*)d_ws;

  const int B = in_sizes[0] / (3 * IMG_H * IMG_W);  // = 8

  dim3 block(TW, TH, 1);
  dim3 grid(IMG_W / TW, IMG_H / TH, B);  // 16 x 64 x 8 = 8192 blocks
  smooth_loss_kernel<<<grid, block, 0, stream>>>(orig, smooth, partial);

  const int nblocks = grid.x * grid.y * grid.z;
  // mean over B * 9 * 3 * H * W elements = 9 * in_sizes[0]
  const double inv_n = 1.0 / (9.0 * (double)in_sizes[0]);
  reduce_partials_kernel<<<1, 256, 0, stream>>>(partial, nblocks, out, inv_n);
}